// CubicBSplineGrid3d_68856915689794
// MI455X (gfx1250) — compile-verified
//
#include <hip/hip_runtime.h>

// Tricubic B-spline interpolation on MI455X (gfx1250, wave32).
//
// Layout: one wave per query point; lane = channel (C == 32 == wave32).
// Every neighbor-row load is a fully coalesced 128B request. Grid (32 MiB)
// is L2-resident (192 MB L2), so this is an L2-gather-bound kernel.
//
// Separable reduction:
//   k-stage + j-stage in VALU (80 of 84 FMAs/channel),
//   i-stage (K=4) via V_WMMA_F32_16X16X4_F32, f32 end-to-end.

typedef __attribute__((ext_vector_type(2))) float v2f;
typedef __attribute__((ext_vector_type(8))) float v8f;

#define GD 64
#define GH 64
#define GW 64
#define GC 32

__device__ __forceinline__ void bspline_axis(float u, int n, int idx[4], float w[4]) {
    // x = u*(n-1); s = floor(x); t = x - s; 4-tap uniform cubic B-spline weights.
    float x = u * (float)(n - 1);
    float s = floorf(x);
    float t = x - s;
    int   si = (int)s;
    float t2 = t * t;
    float t3 = t2 * t;
    const float sixth = 1.0f / 6.0f;
    w[0] = (-t3 + 3.0f * t2 - 3.0f * t + 1.0f) * sixth;
    w[1] = ( 3.0f * t3 - 6.0f * t2 + 4.0f)      * sixth;
    w[2] = (-3.0f * t3 + 3.0f * t2 + 3.0f * t + 1.0f) * sixth;
    w[3] = t3 * sixth;
#pragma unroll
    for (int i = 0; i < 4; ++i) {
        int v = si + i - 1;
        v = v < 0 ? 0 : v;
        v = v > (n - 1) ? (n - 1) : v;
        idx[i] = v;
    }
}

__global__ __launch_bounds__(256) void
tricubic_bspline_kernel(const float* __restrict__ u,
                        const float* __restrict__ grid,
                        float* __restrict__ out, int nq) {
    const int lane = threadIdx.x & 31;   // channel
    const int wid  = threadIdx.x >> 5;   // wave id in block (8 waves/block)
    const int q    = blockIdx.x * 8 + wid;
    if (q >= nq) return;                 // wave-uniform exit; EXEC stays all-1s

    // ---- per-query wave-uniform setup -------------------------------------
    const float* up = u + 3 * q;
    const float ud = up[0];
    const float uh = up[1];
    const float uw = up[2];

    int   id[4], ih[4], iw[4];
    float wd[4], wh[4], ww[4];
    bspline_axis(ud, GD, id, wd);
    bspline_axis(uh, GH, ih, wh);
    bspline_axis(uw, GW, iw, ww);

    int offW[4];
#pragma unroll
    for (int k = 0; k < 4; ++k) offW[k] = iw[k] * GC;

    // ---- gather + k/j contraction in VALU ---------------------------------
    // Q[i][lane] = sum_j wh[j] * sum_k ww[k] * grid[id_i][ih_j][iw_k][lane]
    float Q[4] = {0.0f, 0.0f, 0.0f, 0.0f};
#pragma unroll
    for (int i = 0; i < 4; ++i) {
        const float* pi = grid + (size_t)id[i] * (GH * GW * GC) + lane;
#pragma unroll
        for (int j = 0; j < 4; ++j) {
            const float* pj = pi + ih[j] * (GW * GC);
            // 4 coalesced 128B wave requests (lane = channel)
            float v0 = pj[offW[0]];
            float v1 = pj[offW[1]];
            float v2 = pj[offW[2]];
            float v3 = pj[offW[3]];
            float s = fmaf(v3, ww[3], fmaf(v2, ww[2], fmaf(v1, ww[1], v0 * ww[0])));
            Q[i] = fmaf(wh[j], s, Q[i]);
        }
    }

    // ---- i-stage via V_WMMA_F32_16X16X4_F32 -------------------------------
    // D[m][n] = sum_k A[m][k] * B[k][n]; we use row m=0 only:
    //   A[0][k] = wd[k],  B[k][n] = Q[k][channel n]  (two 16-channel halves).
    //
    // A 16x4 layout (ISA 7.12.2): VGPR0 = K0(lanes0-15)|K2(lanes16-31),
    //                             VGPR1 = K1|K3; lane indexes M.
    // Only lanes 0 (M=0) and 16 (M=0) carry wd; all other lanes zero.
    v2f a;
    a.x = (lane == 0) ? wd[0] : ((lane == 16) ? wd[2] : 0.0f);
    a.y = (lane == 0) ? wd[1] : ((lane == 16) ? wd[3] : 0.0f);

    // B 4x16 layout (rows striped across lanes within a VGPR, mirroring C/D):
    //   VGPR0 = row k=0 (lanes0-15) | row k=2 (lanes16-31)
    //   VGPR1 = row k=1 (lanes0-15) | row k=3 (lanes16-31)
    // Our Q[i] live lane-per-channel; swap wave halves to place them.
    const float Q0s = __shfl_xor(Q[0], 16);
    const float Q1s = __shfl_xor(Q[1], 16);
    const float Q2s = __shfl_xor(Q[2], 16);
    const float Q3s = __shfl_xor(Q[3], 16);
    const bool lo = (lane < 16);

    v2f b0;                         // channels 0..15 tile
    b0.x = lo ? Q[0] : Q2s;
    b0.y = lo ? Q[1] : Q3s;
    v2f b1;                         // channels 16..31 tile
    b1.x = lo ? Q0s : Q[2];
    b1.y = lo ? Q1s : Q[3];

    v8f cz = {};
    v8f dA = __builtin_amdgcn_wmma_f32_16x16x4_f32(
        false, a, false, b0, (short)0, cz, false, false);
    v8f dB = __builtin_amdgcn_wmma_f32_16x16x4_f32(
        false, a, false, b1, (short)0, cz, false, false);

    // D row M=0 sits in VGPR0 lanes 0-15 of each tile (ISA C/D layout).
    // Merge both halves into one full-wave coalesced 128B store.
    const float rHi = __shfl_xor(dB[0], 16);  // lane16+t <- out[ch 16+t]
    const float r   = lo ? dA[0] : rHi;
    __builtin_nontemporal_store(r, out + (size_t)q * GC + lane);
}

extern "C" void kernel_launch(void* const* d_in, const int* in_sizes, int n_in,
                              void* d_out, int out_size, void* d_ws, size_t ws_size,
                              hipStream_t stream) {
    const float* u    = (const float*)d_in[0];   // (b, 3) f32
    const float* grid = (const float*)d_in[1];   // (64, 64, 64, 32) f32
    float*       out  = (float*)d_out;           // (b, 32) f32

    const int nq = in_sizes[0] / 3;              // 131072
    const int waves_per_block = 8;               // 256 threads, wave32
    const int blocks = (nq + waves_per_block - 1) / waves_per_block;

    tricubic_bspline_kernel<<<blocks, 256, 0, stream>>>(u, grid, out, nq);
}